// DGCNNPartSeg_712964571701
// MI455X (gfx1250) — compile-verified
//
#include <hip/hip_runtime.h>
#include <hip/hip_bf16.h>
#include <math.h>

// ---------------------------------------------------------------------------
// DGCNN part-seg forward for MI455X (gfx1250, wave32, WMMA f32 16x16x4).
//
// Edge-conv algebra: W @ concat(nbr - xi, xi) = Bv[:, idx[n,k]] + A[:, n]
// with Bv = W[:, :C] @ x and A = (W[:, C:] - W[:, :C]) @ x, so (B,2C,N,K)
// tensors are never materialized. Inner BN folded into Bv/Aa epilogues.
// GEMMs: branch-free, uniform-base + 32-bit offset addressing (saddr form),
// b64 A-operand loads (packed weights), 4 m-tiles per wave, unroll 8.
// Edge-fused kernel: W slice fully register-resident (full unroll, no movrels).
// ---------------------------------------------------------------------------

typedef __attribute__((ext_vector_type(2))) float v2f;
typedef __attribute__((ext_vector_type(8))) float v8f;

#define NEG_SLOPE 0.2f
#define BN_EPS 1e-5f
#define KNN 40

__device__ __forceinline__ float lrelu(float v) { return v > 0.f ? v : NEG_SLOPE * v; }

// ---------------- BN fold: s = g*rsqrt(v+eps), t = b - m*s -------------------
__global__ void bnprep_kernel(const float* __restrict__ g, const float* __restrict__ b,
                              const float* __restrict__ m, const float* __restrict__ v,
                              float* __restrict__ s, float* __restrict__ t, int C) {
    int c = blockIdx.x * blockDim.x + threadIdx.x;
    if (c < C) {
        float sc = g[c] * rsqrtf(v[c] + BN_EPS);
        s[c] = sc;
        t[c] = b[c] - m[c] * sc;
    }
}

// ---------------- operand prep kernels ---------------------------------------
__global__ void padx_kernel(const float* __restrict__ x, float* __restrict__ x4, int Bn, int N) {
    long tid = blockIdx.x * (long)blockDim.x + threadIdx.x;
    long total = (long)Bn * 4 * N;
    if (tid >= total) return;
    int n = (int)(tid % N);
    int c = (int)((tid / N) % 4);
    long b = tid / (4L * N);
    x4[tid] = (c < 3) ? x[(b * 3 + c) * (long)N + n] : 0.f;
}

__global__ void packW1_kernel(const float* __restrict__ W1, float* __restrict__ W1b,
                              float* __restrict__ W1d) {
    int tid = blockIdx.x * blockDim.x + threadIdx.x;  // 64*4
    if (tid >= 256) return;
    int o = tid >> 2, c = tid & 3;
    float wb = 0.f, wd = 0.f;
    if (c < 3) {
        float w0 = W1[o * 6 + c];
        wb = w0;
        wd = W1[o * 6 + 3 + c] - w0;
    }
    W1b[tid] = wb;
    W1d[tid] = wd;
}

__global__ void packdiff_kernel(const float* __restrict__ W, float* __restrict__ D) {
    int tid = blockIdx.x * blockDim.x + threadIdx.x;  // 64*64, W is (64,128)
    if (tid >= 4096) return;
    int o = tid >> 6, c = tid & 63;
    D[tid] = W[o * 128 + 64 + c] - W[o * 128 + c];
}

__global__ void packW10_kernel(const float* __restrict__ W, float* __restrict__ P) {
    int tid = blockIdx.x * blockDim.x + threadIdx.x;  // 64*128, rows 50..63 zero
    if (tid >= 64 * 128) return;
    int o = tid >> 7;
    P[tid] = (o < 50) ? W[tid] : 0.f;
}

// ---------------- branch-free WMMA f32 GEMM (4 m-tiles / wave) ---------------
// OUT[b,o,m] = act?lrelu:((s? s[o]* : )(sum_c A[o*aOs+c*aCs]*IN[b,c,m]) (+t?t[o]:0))
// Requirements: C % 4 == 0; M % 64 == 0; A readable for 16-aligned padded rows.
// WMMA 16x4 A layout: lane<16 -> (M=lane, K=c0+{0,1}); lane>=16 -> (M=lane-16, K=c0+2+{0,1}).
template <bool ACS1>
__global__ void gemm_wmma_kernel(const float* __restrict__ A, long aBatch, int aOs, int aCs,
                                 const float* __restrict__ IN, long inBatch,
                                 const float* __restrict__ s, const float* __restrict__ t, int act,
                                 float* __restrict__ OUT, long outBatch,
                                 int O, int C, int M, int Bn) {
    int gtid = blockIdx.x * blockDim.x + threadIdx.x;
    int wave = gtid >> 5;
    int lane = gtid & 31;
    int otiles = (O + 15) >> 4;
    int mtiles = M >> 6;  // four 16-col tiles per wave
    long tilesPerB = (long)otiles * mtiles;
    long bIdx = wave / tilesPerB;
    if (bIdx >= Bn) return;  // uniform per wave -> WMMA runs with full EXEC
    int tb = (int)(wave % tilesPerB);
    int ot = tb / mtiles, mt = tb % mtiles;
    int hi = lane >> 4, lo = lane & 15;
    int row = ot * 16 + lo;
    int col = mt * 64 + lo;
    const float* Ab = A + bIdx * aBatch;   // uniform base (saddr)
    const float* Ib = IN + bIdx * inBatch; // uniform base (saddr)
    int aoff = row * aOs + 2 * hi * aCs;   // 32-bit lane offsets
    int boff = 2 * hi * M + col;
    const int aStep = 4 * aCs;
    const int bStep = 4 * M;
    v8f acc0 = {0.f, 0.f, 0.f, 0.f, 0.f, 0.f, 0.f, 0.f};
    v8f acc1 = acc0, acc2 = acc0, acc3 = acc0;
    int iters = C >> 2;
#pragma unroll 8
    for (int it = 0; it < iters; ++it) {
        v2f av;
        if (ACS1) {
            av = *(const v2f*)(Ab + aoff);  // contiguous pair -> global_load_b64
        } else {
            av.x = Ab[aoff];
            av.y = Ab[aoff + aCs];
        }
        v2f b0; b0.x = Ib[boff];      b0.y = Ib[boff + M];
        v2f b1; b1.x = Ib[boff + 16]; b1.y = Ib[boff + M + 16];
        v2f b2; b2.x = Ib[boff + 32]; b2.y = Ib[boff + M + 32];
        v2f b3; b3.x = Ib[boff + 48]; b3.y = Ib[boff + M + 48];
        acc0 = __builtin_amdgcn_wmma_f32_16x16x4_f32(false, av, false, b0, (short)0, acc0,
                                                     false, false);
        acc1 = __builtin_amdgcn_wmma_f32_16x16x4_f32(false, av, false, b1, (short)0, acc1,
                                                     false, false);
        acc2 = __builtin_amdgcn_wmma_f32_16x16x4_f32(false, av, false, b2, (short)0, acc2,
                                                     false, false);
        acc3 = __builtin_amdgcn_wmma_f32_16x16x4_f32(false, av, false, b3, (short)0, acc3,
                                                     false, false);
        aoff += aStep;
        boff += bStep;
    }
    long obase = bIdx * outBatch + col;
#pragma unroll
    for (int i = 0; i < 8; ++i) {
        int r = ot * 16 + i + 8 * hi;  // C/D layout: VGPR i -> M=i (lanes 0-15) / M=i+8
        if (r < O) {
            float v0 = acc0[i], v1 = acc1[i], v2 = acc2[i], v3 = acc3[i];
            if (s) { float sr = s[r]; v0 *= sr; v1 *= sr; v2 *= sr; v3 *= sr; }
            if (t) { float tr = t[r]; v0 += tr; v1 += tr; v2 += tr; v3 += tr; }
            if (act) { v0 = lrelu(v0); v1 = lrelu(v1); v2 = lrelu(v2); v3 = lrelu(v3); }
            float* op = OUT + obase + (long)r * M;
            op[0] = v0;
            op[16] = v1;
            op[32] = v2;
            op[48] = v3;
        }
    }
}

// ---------------- top-40 on pd[n,m] = 2G[n,m] - G[n,n] - G[m,m] --------------
__global__ void topk_kernel(const float* __restrict__ G, int* __restrict__ idxOut, int N) {
    __shared__ float diag[2048];
    for (int j = threadIdx.x; j < N; j += blockDim.x) diag[j] = G[(long)j * N + j];
    __syncthreads();
    int n = blockIdx.x * blockDim.x + threadIdx.x;
    if (n >= N) return;
    float vals[KNN];
    int inds[KNN];
    for (int j = 0; j < KNN; ++j) { vals[j] = -INFINITY; inds[j] = 0; }
    const float* row = G + (long)n * N;
    float dn = diag[n];
    for (int m = 0; m < N; ++m) {
        float v = 2.f * row[m] - dn - diag[m];
        if (v > vals[KNN - 1]) {  // strict: ties keep lower index, like lax.top_k
            int p = KNN - 1;
            while (p > 0 && vals[p - 1] < v) {
                vals[p] = vals[p - 1];
                inds[p] = inds[p - 1];
                --p;
            }
            vals[p] = v;
            inds[p] = m;
        }
    }
    int* op = idxOut + (long)n * KNN;
    for (int j = 0; j < KNN; ++j) op[j] = inds[j];
}

// ---- fused: h[c] = lrelu(Bv'[c,idx] + Aa'[c,n]) (BN prefolded) as WMMA B ----
// ---- operand -> W(64x64) -> bn_out+lrelu -> max over K=40. Wave/(b,ot,n). ---
// Single pass: 3 column-tiles (40 cols) in 3 accumulators; Aa loads shared.
// Fully unrolled so the 16 W pairs stay as direct register operands.
__global__ void edge_fused_max_kernel(const float* __restrict__ Bv, const float* __restrict__ Aa,
                                      const int* __restrict__ idx,
                                      const float* __restrict__ W,
                                      const float* __restrict__ so, const float* __restrict__ to,
                                      float* __restrict__ Xout, int Bn, int N, int Kn) {
    int gtid = blockIdx.x * blockDim.x + threadIdx.x;
    int wave = gtid >> 5;
    int lane = gtid & 31;
    long per = 4L * N;  // 4 o-tiles (O=64)
    long b = wave / per;
    if (b >= Bn) return;
    int r0 = (int)(wave % per);
    int ot = r0 / N;
    int n = r0 % N;
    int hi = lane >> 4, lo = lane & 15;
    // preload this lane's 16 W pairs (b64 loads)
    v2f wreg[16];
    {
        const float* wp = W + (ot * 16 + lo) * 64 + 2 * hi;
#pragma unroll
        for (int j = 0; j < 16; ++j) wreg[j] = *(const v2f*)(wp + 4 * j);
    }
    const float* Bvb = Bv + b * 64L * N;  // uniform bases
    const float* Ab = Aa + b * 64L * N;
    const int* idxb = idx + (b * (long)N + n) * Kn;
    bool valid2 = lo < (Kn - 32);  // third tile covers cols 32..47, Kn=40
    int gi0 = idxb[lo];
    int gi1 = idxb[16 + lo];
    int gi2 = idxb[valid2 ? (32 + lo) : 0];
    int base = 2 * hi * N;
    int o0 = base + gi0, o1 = base + gi1, o2 = base + gi2, oa = base + n;
    v8f a0 = {0.f, 0.f, 0.f, 0.f, 0.f, 0.f, 0.f, 0.f};
    v8f a1 = a0, a2 = a0;
    const int step = 4 * N;
#pragma unroll
    for (int j = 0; j < 16; ++j) {
        float aaL = Ab[oa], aaH = Ab[oa + N];
        v2f x0; x0.x = lrelu(Bvb[o0] + aaL); x0.y = lrelu(Bvb[o0 + N] + aaH);
        v2f x1; x1.x = lrelu(Bvb[o1] + aaL); x1.y = lrelu(Bvb[o1 + N] + aaH);
        v2f x2; x2.x = lrelu(Bvb[o2] + aaL); x2.y = lrelu(Bvb[o2 + N] + aaH);
        v2f av = wreg[j];
        a0 = __builtin_amdgcn_wmma_f32_16x16x4_f32(false, av, false, x0, (short)0, a0,
                                                   false, false);
        a1 = __builtin_amdgcn_wmma_f32_16x16x4_f32(false, av, false, x1, (short)0, a1,
                                                   false, false);
        a2 = __builtin_amdgcn_wmma_f32_16x16x4_f32(false, av, false, x2, (short)0, a2,
                                                   false, false);
        o0 += step; o1 += step; o2 += step; oa += step;
    }
    long xbase = b * 64L * N + n;
#pragma unroll
    for (int i = 0; i < 8; ++i) {
        int r = ot * 16 + i + 8 * hi;
        float sr = so[r], tr = to[r];
        float v0 = lrelu(sr * a0[i] + tr);
        float v1 = lrelu(sr * a1[i] + tr);
        float v2 = valid2 ? lrelu(sr * a2[i] + tr) : -INFINITY;
        float v = fmaxf(fmaxf(v0, v1), v2);
        for (int msk = 1; msk < 16; msk <<= 1) v = fmaxf(v, __shfl_xor(v, msk, 32));
        if (lo == 0) Xout[xbase + (long)r * N] = v;  // lane0: rows 0..7, lane16: rows 8..15
    }
}

// ---------------- conv5 (BN prefolded): x3 = max_k lrelu(Bv'+Aa') ------------
__global__ void edge_conv_max_kernel(const float* __restrict__ Bv, const float* __restrict__ Aa,
                                     const int* __restrict__ idx, float* __restrict__ Xout,
                                     int Bn, int N, int Kn) {
    long tid = blockIdx.x * (long)blockDim.x + threadIdx.x;
    long total = (long)Bn * 64 * N;
    if (tid >= total) return;
    int n = (int)(tid % N);
    int r = (int)((tid / N) % 64);
    long b = tid / (64L * N);
    const float* Bvr = Bv + b * 64L * N + (long)r * N;
    float av = Aa[b * 64L * N + (long)r * N + n];
    const int* ib = idx + (b * (long)N + n) * Kn;
    float best = -INFINITY;
    for (int k = 0; k < Kn; ++k) {
        best = fmaxf(best, lrelu(Bvr[ib[k]] + av));
    }
    Xout[tid] = best;
}

// ---------------- cls branch: (B,64) = lrelu(bn(Wcls @ one_hot)) -------------
__global__ void cls_kernel(const float* __restrict__ Wc, const float* __restrict__ lab,
                           const float* __restrict__ s, const float* __restrict__ t,
                           float* __restrict__ out, int Bn) {
    int tid = blockIdx.x * blockDim.x + threadIdx.x;
    if (tid >= Bn * 64) return;
    int b = tid / 64, o = tid % 64;
    float acc = 0.f;
    for (int c = 0; c < 16; ++c) acc = fmaf(Wc[o * 16 + c], lab[b * 16 + c], acc);
    out[tid] = lrelu(s[o] * acc + t[o]);
}

// ---------------- max over N (global pooling) --------------------------------
__global__ void maxn_kernel(const float* __restrict__ H, float* __restrict__ out, int total,
                            int N) {
    int tid = blockIdx.x * blockDim.x + threadIdx.x;
    if (tid >= total) return;
    const float* p = H + (long)tid * N;
    float best = -INFINITY;
    for (int n = 0; n < N; ++n) best = fmaxf(best, p[n]);
    out[tid] = best;
}

// ---------------- channel concats -------------------------------------------
__global__ void pack192_kernel(const float* __restrict__ x1, const float* __restrict__ x2,
                               const float* __restrict__ x3, float* __restrict__ cat, int Bn,
                               int N) {
    long tid = blockIdx.x * (long)blockDim.x + threadIdx.x;
    long total = (long)Bn * 192 * N;
    if (tid >= total) return;
    int n = (int)(tid % N);
    int c = (int)((tid / N) % 192);
    long b = tid / (192L * N);
    const float* src;
    if (c < 64)       src = x1 + (b * 64 + c) * (long)N;
    else if (c < 128) src = x2 + (b * 64 + (c - 64)) * (long)N;
    else              src = x3 + (b * 64 + (c - 128)) * (long)N;
    cat[tid] = src[n];
}

__global__ void pack1280_kernel(const float* __restrict__ xg, const float* __restrict__ x1,
                                const float* __restrict__ x2, const float* __restrict__ x3,
                                const float* __restrict__ clsf, float* __restrict__ cat, int Bn,
                                int N) {
    long tid = blockIdx.x * (long)blockDim.x + threadIdx.x;
    long total = (long)Bn * 1280 * N;
    if (tid >= total) return;
    int n = (int)(tid % N);
    int c = (int)((tid / N) % 1280);
    long b = tid / (1280L * N);
    float v;
    if (c < 1024)      v = xg[b * 1024 + c];
    else if (c < 1088) v = x1[(b * 64 + (c - 1024)) * (long)N + n];
    else if (c < 1152) v = x2[(b * 64 + (c - 1088)) * (long)N + n];
    else if (c < 1216) v = x3[(b * 64 + (c - 1152)) * (long)N + n];
    else               v = clsf[b * 64 + (c - 1216)];
    cat[tid] = v;
}

// ---------------------------------------------------------------------------
extern "C" void kernel_launch(void* const* d_in, const int* in_sizes, int n_in,
                              void* d_out, int out_size, void* d_ws, size_t ws_size,
                              hipStream_t stream) {
    (void)in_sizes; (void)n_in; (void)out_size; (void)ws_size;
    const int B = 8, N = 2048, K = KNN;

    // Inputs in setup_inputs() dict order (bn tuples flatten as g,b,m,v):
    const float* x    = (const float*)d_in[0];
    const float* cls  = (const float*)d_in[1];
    const float* W1   = (const float*)d_in[2];
    const float* W2   = (const float*)d_in[7];
    const float* W3   = (const float*)d_in[12];
    const float* W4   = (const float*)d_in[17];
    const float* W5   = (const float*)d_in[22];
    const float* W6   = (const float*)d_in[27];
    const float* W7   = (const float*)d_in[32];
    const float* W8   = (const float*)d_in[37];
    const float* W9   = (const float*)d_in[42];
    const float* W10  = (const float*)d_in[47];
    const float* b10  = (const float*)d_in[48];
    const float* Wcls = (const float*)d_in[49];
    // s/t index: 0..4 = bn1..bn5, 5 = bn6, 6 = bn7, 7 = bn8, 8 = bn9, 9 = bncls
    const int bnBase[10] = {3, 8, 13, 18, 23, 28, 33, 38, 43, 50};
    const int bnC[10]    = {64, 64, 64, 64, 64, 1024, 256, 256, 128, 64};

    // ---------------- workspace layout ----------------
    char* wsb = (char*)d_ws;
    size_t off = 0;
    auto alloc = [&](size_t bytes) -> void* {
        void* p = wsb + off;
        off += (bytes + 255) & ~(size_t)255;
        return p;
    };
    float *sArr[10], *tArr[10];
    for (int i = 0; i < 10; ++i) {
        sArr[i] = (float*)alloc((size_t)bnC[i] * 4);
        tArr[i] = (float*)alloc((size_t)bnC[i] * 4);
    }
    float* x4      = (float*)alloc((size_t)B * 4 * N * 4);
    float* W1b     = (float*)alloc(256 * 4);
    float* W1d     = (float*)alloc(256 * 4);
    float* W3d     = (float*)alloc(4096 * 4);
    float* W5d     = (float*)alloc(4096 * 4);
    float* W10p    = (float*)alloc(64 * 128 * 4);
    int*   idxBuf  = (int*)alloc((size_t)B * N * K * 4);
    float* Gbuf    = (float*)alloc((size_t)N * N * 4);  // per-batch Gram, reused
    float* BvBuf   = (float*)alloc((size_t)B * 64 * N * 4);
    float* AaBuf   = (float*)alloc((size_t)B * 64 * N * 4);
    float* x1Buf   = (float*)alloc((size_t)B * 64 * N * 4);
    float* x2Buf   = (float*)alloc((size_t)B * 64 * N * 4);
    float* x3Buf   = (float*)alloc((size_t)B * 64 * N * 4);
    float* cat192  = (float*)alloc((size_t)B * 192 * N * 4);
    float* h6      = (float*)alloc((size_t)B * 1024 * N * 4);
    float* xg      = (float*)alloc((size_t)B * 1024 * 4);
    float* clsf    = (float*)alloc((size_t)B * 64 * 4);
    float* cat1280 = (float*)alloc((size_t)B * 1280 * N * 4);
    float* h7      = (float*)alloc((size_t)B * 256 * N * 4);
    float* h8      = (float*)alloc((size_t)B * 256 * N * 4);
    float* h9      = (float*)alloc((size_t)B * 128 * N * 4);

    // ---------------- prep: BN fold, operand packs, cls ----------------
    for (int i = 0; i < 10; ++i) {
        bnprep_kernel<<<(bnC[i] + 255) / 256, 256, 0, stream>>>(
            (const float*)d_in[bnBase[i] + 0], (const float*)d_in[bnBase[i] + 1],
            (const float*)d_in[bnBase[i] + 2], (const float*)d_in[bnBase[i] + 3],
            sArr[i], tArr[i], bnC[i]);
    }
    padx_kernel<<<(int)(((long)B * 4 * N + 255) / 256), 256, 0, stream>>>(x, x4, B, N);
    packW1_kernel<<<1, 256, 0, stream>>>(W1, W1b, W1d);
    packdiff_kernel<<<16, 256, 0, stream>>>(W3, W3d);
    packdiff_kernel<<<16, 256, 0, stream>>>(W5, W5d);
    packW10_kernel<<<32, 256, 0, stream>>>(W10, W10p);
    cls_kernel<<<(B * 64 + 255) / 256, 256, 0, stream>>>(Wcls, cls, sArr[9], tArr[9], clsf, B);

    auto gemm = [&](const float* A, long aBatch, int aOs, int aCs, const float* IN, long inBatch,
                    const float* s, const float* t, int act, float* OUT, long outBatch, int O,
                    int C, int M, int Bn) {
        long waves = (long)Bn * ((O + 15) / 16) * (M / 64);
        int blocks = (int)((waves * 32 + 255) / 256);
        if (aCs == 1)
            gemm_wmma_kernel<true><<<blocks, 256, 0, stream>>>(A, aBatch, aOs, aCs, IN, inBatch,
                                                               s, t, act, OUT, outBatch, O, C, M,
                                                               Bn);
        else
            gemm_wmma_kernel<false><<<blocks, 256, 0, stream>>>(A, aBatch, aOs, aCs, IN, inBatch,
                                                                s, t, act, OUT, outBatch, O, C, M,
                                                                Bn);
    };
    auto knn = [&](const float* pts, int C) {  // pts: (B,C,N); A reads X^T via strides
        for (int b = 0; b < B; ++b) {
            const float* xb = pts + (long)b * C * N;
            gemm(xb, 0, /*aOs=*/1, /*aCs=*/N, xb, 0, nullptr, nullptr, 0, Gbuf, 0, N, C, N, 1);
            topk_kernel<<<N / 256, 256, 0, stream>>>(Gbuf, idxBuf + (long)b * N * K, N);
        }
    };
    auto edgeFused = [&](const float* W, const float* so, const float* to, float* out) {
        long waves = (long)B * 4 * N;
        int blocks = (int)((waves * 32 + 255) / 256);
        edge_fused_max_kernel<<<blocks, 256, 0, stream>>>(BvBuf, AaBuf, idxBuf, W, so, to, out,
                                                          B, N, K);
    };

    // ---- stage 1: x -> knn -> conv1 (folded bn1) -> conv2+bn2 -> max -------
    knn(x4, 4);
    gemm(W1b, 0, 4, 1, x4, 4L * N, sArr[0], nullptr, 0, BvBuf, 64L * N, 64, 4, N, B);
    gemm(W1d, 0, 4, 1, x4, 4L * N, sArr[0], tArr[0], 0, AaBuf, 64L * N, 64, 4, N, B);
    edgeFused(W2, sArr[1], tArr[1], x1Buf);

    // ---- stage 2: x1 -> knn -> conv3 (folded bn3) -> conv4+bn4 -> max ------
    knn(x1Buf, 64);
    gemm(W3, 0, 128, 1, x1Buf, 64L * N, sArr[2], nullptr, 0, BvBuf, 64L * N, 64, 64, N, B);
    gemm(W3d, 0, 64, 1, x1Buf, 64L * N, sArr[2], tArr[2], 0, AaBuf, 64L * N, 64, 64, N, B);
    edgeFused(W4, sArr[3], tArr[3], x2Buf);

    // ---- stage 3: x2 -> knn -> conv5 (folded bn5) -> max -------------------
    knn(x2Buf, 64);
    gemm(W5, 0, 128, 1, x2Buf, 64L * N, sArr[4], nullptr, 0, BvBuf, 64L * N, 64, 64, N, B);
    gemm(W5d, 0, 64, 1, x2Buf, 64L * N, sArr[4], tArr[4], 0, AaBuf, 64L * N, 64, 64, N, B);
    {
        long total = (long)B * 64 * N;
        edge_conv_max_kernel<<<(int)((total + 255) / 256), 256, 0, stream>>>(
            BvBuf, AaBuf, idxBuf, x3Buf, B, N, K);
    }

    // ---------------- global branch -----------------------------------------
    {
        long total = (long)B * 192 * N;
        pack192_kernel<<<(int)((total + 255) / 256), 256, 0, stream>>>(x1Buf, x2Buf, x3Buf,
                                                                       cat192, B, N);
    }
    gemm(W6, 0, 192, 1, cat192, 192L * N, sArr[5], tArr[5], 1, h6, 1024L * N, 1024, 192, N, B);
    maxn_kernel<<<(B * 1024 + 255) / 256, 256, 0, stream>>>(h6, xg, B * 1024, N);
    {
        long total = (long)B * 1280 * N;
        pack1280_kernel<<<(int)((total + 255) / 256), 256, 0, stream>>>(xg, x1Buf, x2Buf, x3Buf,
                                                                        clsf, cat1280, B, N);
    }

    // ---------------- head --------------------------------------------------
    gemm(W7, 0, 1280, 1, cat1280, 1280L * N, sArr[6], tArr[6], 1, h7, 256L * N, 256, 1280, N, B);
    gemm(W8, 0, 256, 1, h7, 256L * N, sArr[7], tArr[7], 1, h8, 256L * N, 256, 256, N, B);
    gemm(W9, 0, 256, 1, h8, 256L * N, sArr[8], tArr[8], 1, h9, 128L * N, 128, 256, N, B);
    gemm(W10p, 0, 128, 1, h9, 128L * N, nullptr, b10, 0, (float*)d_out, 50L * N, 50, 128, N, B);
}